// GCNModel_1855425872413
// MI455X (gfx1250) — compile-verified
//
#include <hip/hip_runtime.h>
#include <hip/hip_bf16.h>
#include <math.h>

// ---------------- problem constants (match reference) ----------------
#define N_USER 50000
#define N_ITEM 30000
#define NTOT   (N_USER + N_ITEM)   // 80000
#define D      64
#define NNZ_UI 1000000
#define NNZ_R  800000
#define NNZ_II 300000
#define IMG_F  512
#define TXT_F  384
#define EPSBN  1e-5f

typedef __attribute__((ext_vector_type(2))) float v2f;
typedef __attribute__((ext_vector_type(8))) float v8f;

// ---------------------------------------------------------------------
// Build cat = concat(item_emb, user_emb)  -> [NTOT, 64]
// ---------------------------------------------------------------------
__global__ void build_cat_kernel(const float* __restrict__ item_emb,
                                 const float* __restrict__ user_emb,
                                 float* __restrict__ cat)
{
    size_t idx = (size_t)blockIdx.x * blockDim.x + threadIdx.x;
    size_t total = (size_t)NTOT * D;
    if (idx >= total) return;
    size_t item_part = (size_t)N_ITEM * D;
    cat[idx] = (idx < item_part) ? item_emb[idx] : user_emb[idx - item_part];
}

// ---------------------------------------------------------------------
// SpMM via edge-parallel atomics: y[rows[e]] += vals[e] * x[cols[e]]
// 16 threads per edge, one float4 chunk each (D=64).
// y must be pre-zeroed.
// ---------------------------------------------------------------------
__global__ void spmm_atomic_kernel(const int* __restrict__ rows,
                                   const int* __restrict__ cols,
                                   const float* __restrict__ vals,
                                   const float* __restrict__ x,
                                   float* __restrict__ y,
                                   int nnz)
{
    long long total = (long long)nnz * 16;
    long long stride = (long long)gridDim.x * blockDim.x;
    for (long long t = (long long)blockIdx.x * blockDim.x + threadIdx.x;
         t < total; t += stride) {
        int e = (int)(t >> 4);
        int c = (int)(t & 15);
        int r  = rows[e];
        int cl = cols[e];
        float v = vals[e];
        const float4 xv = ((const float4*)(x + (size_t)cl * D))[c];
        float* yp = y + (size_t)r * D + (size_t)c * 4;
        atomicAdd(yp + 0, v * xv.x);
        atomicAdd(yp + 1, v * xv.y);
        atomicAdd(yp + 2, v * xv.z);
        atomicAdd(yp + 3, v * xv.w);
    }
}

// ---------------------------------------------------------------------
// content = (c0 + c1 + c2) / 3
// ---------------------------------------------------------------------
__global__ void mean3_kernel(const float* __restrict__ c0,
                             const float* __restrict__ c1,
                             const float* __restrict__ c2,
                             float* __restrict__ o, size_t n)
{
    size_t idx = (size_t)blockIdx.x * blockDim.x + threadIdx.x;
    if (idx >= n) return;
    o[idx] = (c0[idx] + c1[idx] + c2[idx]) * (1.0f / 3.0f);
}

// ---------------------------------------------------------------------
// WMMA f32 GEMM:  Y[M,64] = epilogue( X[M,K] @ W[K,64] )
//   mode 0: h = (h+bias)*gamma*rsqrt(1+eps)+beta ; leaky_relu      (projection)
//   mode 1: h = h+bias ; leaky_relu                                 (attention)
//   mode 2: h = aux * sigmoid(h+bias)                               (gate)
// One wave per 16x16 output tile, K-loop of v_wmma_f32_16x16x4_f32.
// A layout (16x4 f32): lanes 0-15 rows, VGPR0={K0,K2}, VGPR1={K1,K3}
// B layout (4x16 f32): lanes 0-15 cols, VGPR0={K0,K2}, VGPR1={K1,K3}
// C/D layout: VGPR r -> M=r (lanes 0-15) / M=r+8 (lanes 16-31), N=lane&15
// ---------------------------------------------------------------------
__global__ void gemm64_wmma_kernel(const float* __restrict__ X,
                                   const float* __restrict__ W,
                                   const float* __restrict__ bias,
                                   const float* __restrict__ gamma,
                                   const float* __restrict__ beta,
                                   const float* __restrict__ aux,
                                   float* __restrict__ Y,
                                   int M, int K, int mode)
{
    const int wave = (int)((blockIdx.x * blockDim.x + threadIdx.x) >> 5);
    const int lane = threadIdx.x & 31;
    const int tiles_n = D / 16;           // 4
    const int tile_m = wave / tiles_n;
    const int tile_n = wave % tiles_n;
    if (tile_m * 16 >= M) return;         // wave-uniform: EXEC stays all-1s

    const int half = lane >> 4;           // 0: K0/K1  1: K2/K3
    const int l    = lane & 15;
    const int arow = tile_m * 16 + l;
    const int bcol = tile_n * 16 + l;

    v8f c = {};
    const float* xr = X + (size_t)arow * K;
    for (int k = 0; k < K; k += 4) {
        const int kk = k + half * 2;
        v2f a;
        a.x = xr[kk];
        a.y = xr[kk + 1];
        v2f b;
        b.x = W[(size_t)kk * D + bcol];
        b.y = W[(size_t)(kk + 1) * D + bcol];
        c = __builtin_amdgcn_wmma_f32_16x16x4_f32(
                /*neg_a=*/false, a, /*neg_b=*/false, b,
                /*c_mod=*/(short)0, c, /*reuse_a=*/false, /*reuse_b=*/false);
    }

    const float bn_inv = rsqrtf(1.0f + EPSBN);
    const int col = tile_n * 16 + l;
    const float bo = bias[col];
    #pragma unroll
    for (int r = 0; r < 8; ++r) {
        const int row = tile_m * 16 + r + half * 8;
        float h = c[r] + bo;
        if (mode == 0) {
            h = h * (gamma[col] * bn_inv) + beta[col];
            h = (h >= 0.0f) ? h : 0.01f * h;
        } else if (mode == 1) {
            h = (h >= 0.0f) ? h : 0.01f * h;
        } else {
            const float s = 1.0f / (1.0f + __expf(-h));
            h = aux[(size_t)row * D + col] * s;
        }
        Y[(size_t)row * D + col] = h;
    }
}

// ---------------------------------------------------------------------
// copy n floats
// ---------------------------------------------------------------------
__global__ void copy_kernel(const float* __restrict__ src,
                            float* __restrict__ dst, size_t n)
{
    size_t idx = (size_t)blockIdx.x * blockDim.x + threadIdx.x;
    if (idx >= n) return;
    dst[idx] = src[idx];
}

// ---------------------------------------------------------------------
// s[row] = dot(h[row, 0:64], w2[0:64])  -- one wave per row
// ---------------------------------------------------------------------
__global__ void rowdot_kernel(const float* __restrict__ h,
                              const float* __restrict__ w2,
                              float* __restrict__ s, int nrows)
{
    int wave = (int)((blockIdx.x * blockDim.x + threadIdx.x) >> 5);
    int lane = threadIdx.x & 31;
    if (wave >= nrows) return;
    const float* hr = h + (size_t)wave * D;
    float p = hr[lane] * w2[lane] + hr[lane + 32] * w2[lane + 32];
    #pragma unroll
    for (int off = 16; off > 0; off >>= 1)
        p += __shfl_xor(p, off, 32);
    if (lane == 0) s[wave] = p;
}

// ---------------------------------------------------------------------
// softmax over 2 scores, fuse -> special_img, special_txt, common
// ---------------------------------------------------------------------
__global__ void fuse_kernel(const float* __restrict__ image_ui,
                            const float* __restrict__ text_ui,
                            const float* __restrict__ s_img,
                            const float* __restrict__ s_txt,
                            float* __restrict__ out_sp_img,
                            float* __restrict__ out_sp_txt,
                            float* __restrict__ out_common,
                            size_t n)
{
    size_t idx = (size_t)blockIdx.x * blockDim.x + threadIdx.x;
    if (idx >= n) return;
    size_t row = idx >> 6;
    float a = s_img[row], b = s_txt[row];
    float m = fmaxf(a, b);
    float ea = __expf(a - m), eb = __expf(b - m);
    float inv = 1.0f / (ea + eb);
    float w0 = ea * inv, w1 = eb * inv;
    float ei = image_ui[idx], et = text_ui[idx];
    float com = w0 * ei + w1 * et;
    out_sp_img[idx] = ei - com;
    out_sp_txt[idx] = et - com;
    out_common[idx] = com;
}

// ---------------------------------------------------------------------
// host-side launcher
// ---------------------------------------------------------------------
static inline int blocks_for(size_t n, int tpb) { return (int)((n + tpb - 1) / tpb); }

extern "C" void kernel_launch(void* const* d_in, const int* in_sizes, int n_in,
                              void* d_out, int out_size, void* d_ws, size_t ws_size,
                              hipStream_t stream)
{
    (void)in_sizes; (void)n_in; (void)out_size; (void)ws_size;

    const float* user_emb   = (const float*)d_in[0];
    const float* item_emb   = (const float*)d_in[1];
    const float* image_fe   = (const float*)d_in[2];
    const float* text_fe    = (const float*)d_in[3];
    const float* W_img      = (const float*)d_in[4];
    const float* b_img      = (const float*)d_in[5];
    const float* g_img      = (const float*)d_in[6];
    const float* be_img     = (const float*)d_in[7];
    const float* W_txt      = (const float*)d_in[8];
    const float* b_txt      = (const float*)d_in[9];
    const float* g_txt      = (const float*)d_in[10];
    const float* be_txt     = (const float*)d_in[11];
    const float* W_gate_img = (const float*)d_in[12];
    const float* b_gate_img = (const float*)d_in[13];
    const float* W_gate_txt = (const float*)d_in[14];
    const float* b_gate_txt = (const float*)d_in[15];
    const float* W_common1  = (const float*)d_in[16];
    const float* b_common1  = (const float*)d_in[17];
    const float* w_common2  = (const float*)d_in[18];
    const float* ui_vals    = (const float*)d_in[19];
    const float* R_vals     = (const float*)d_in[20];
    const float* ii_i_vals  = (const float*)d_in[21];
    const float* ii_t_vals  = (const float*)d_in[22];
    const int*   ui_rows    = (const int*)d_in[23];
    const int*   ui_cols    = (const int*)d_in[24];
    const int*   R_rows     = (const int*)d_in[25];
    const int*   R_cols     = (const int*)d_in[26];
    const int*   ii_i_rows  = (const int*)d_in[27];
    const int*   ii_i_cols  = (const int*)d_in[28];
    const int*   ii_t_rows  = (const int*)d_in[29];
    const int*   ii_t_cols  = (const int*)d_in[30];

    float* out = (float*)d_out;

    const size_t ND = (size_t)NTOT * D;     // 5,120,000
    const size_t ID = (size_t)N_ITEM * D;   // 1,920,000
    const size_t UD = (size_t)N_USER * D;   // 3,200,000

    float* ws = (float*)d_ws;
    float* cat0     = ws;                   // [NTOT,D]  later: image_ui
    float* cat1     = cat0 + ND;            // [NTOT,D]  later: text_ui
    float* cat2     = cat1 + ND;            // [NTOT,D]  later: h
    float* img_feat = cat2 + ND;            // [N_ITEM,D]
    float* txt_feat = img_feat + ID;        // [N_ITEM,D]
    float* itA      = txt_feat + ID;        // [N_ITEM,D]
    float* itB      = itA + ID;             // [N_ITEM,D]
    float* s_img    = itB + ID;             // [NTOT]
    float* s_txt    = s_img + NTOT;         // [NTOT]
    float* image_ui = cat0;
    float* text_ui  = cat1;
    float* hbuf     = cat2;

    const int TPB = 256;

    // ---- user_item GCN: cat0 -> cat1 -> cat2, mean into out[0] ----
    build_cat_kernel<<<blocks_for(ND, TPB), TPB, 0, stream>>>(item_emb, user_emb, cat0);

    hipMemsetAsync(cat1, 0, ND * sizeof(float), stream);
    spmm_atomic_kernel<<<blocks_for((size_t)NNZ_UI * 16, TPB), TPB, 0, stream>>>(
        ui_rows, ui_cols, ui_vals, cat0, cat1, NNZ_UI);

    hipMemsetAsync(cat2, 0, ND * sizeof(float), stream);
    spmm_atomic_kernel<<<blocks_for((size_t)NNZ_UI * 16, TPB), TPB, 0, stream>>>(
        ui_rows, ui_cols, ui_vals, cat1, cat2, NNZ_UI);

    mean3_kernel<<<blocks_for(ND, TPB), TPB, 0, stream>>>(cat0, cat1, cat2, out, ND);

    // ---- modal projections (WMMA f32 GEMM + bias + BN + leaky) ----
    {
        int waves = ((N_ITEM + 15) / 16) * (D / 16);
        int blk = blocks_for((size_t)waves * 32, TPB);
        gemm64_wmma_kernel<<<blk, TPB, 0, stream>>>(
            image_fe, W_img, b_img, g_img, be_img, nullptr, img_feat, N_ITEM, IMG_F, 0);
        gemm64_wmma_kernel<<<blk, TPB, 0, stream>>>(
            text_fe, W_txt, b_txt, g_txt, be_txt, nullptr, txt_feat, N_ITEM, TXT_F, 0);
    }

    // ---- modal branch helper (inline, image then text) ----
    const int gate_blk = blocks_for((size_t)((N_ITEM + 15) / 16) * (D / 16) * 32, TPB);
    const int ii_blk   = blocks_for((size_t)NNZ_II * 16, TPB);
    const int R_blk    = blocks_for((size_t)NNZ_R * 16, TPB);

    // image branch
    gemm64_wmma_kernel<<<gate_blk, TPB, 0, stream>>>(
        img_feat, W_gate_img, b_gate_img, nullptr, nullptr, item_emb, itA, N_ITEM, D, 2);
    hipMemsetAsync(itB, 0, ID * sizeof(float), stream);
    spmm_atomic_kernel<<<ii_blk, TPB, 0, stream>>>(ii_i_rows, ii_i_cols, ii_i_vals, itA, itB, NNZ_II);
    hipMemsetAsync(itA, 0, ID * sizeof(float), stream);
    spmm_atomic_kernel<<<ii_blk, TPB, 0, stream>>>(ii_i_rows, ii_i_cols, ii_i_vals, itB, itA, NNZ_II);
    hipMemsetAsync(image_ui, 0, UD * sizeof(float), stream);
    spmm_atomic_kernel<<<R_blk, TPB, 0, stream>>>(R_rows, R_cols, R_vals, itA, image_ui, NNZ_R);
    copy_kernel<<<blocks_for(ID, TPB), TPB, 0, stream>>>(itA, image_ui + UD, ID);

    // text branch
    gemm64_wmma_kernel<<<gate_blk, TPB, 0, stream>>>(
        txt_feat, W_gate_txt, b_gate_txt, nullptr, nullptr, item_emb, itA, N_ITEM, D, 2);
    hipMemsetAsync(itB, 0, ID * sizeof(float), stream);
    spmm_atomic_kernel<<<ii_blk, TPB, 0, stream>>>(ii_t_rows, ii_t_cols, ii_t_vals, itA, itB, NNZ_II);
    hipMemsetAsync(itA, 0, ID * sizeof(float), stream);
    spmm_atomic_kernel<<<ii_blk, TPB, 0, stream>>>(ii_t_rows, ii_t_cols, ii_t_vals, itB, itA, NNZ_II);
    hipMemsetAsync(text_ui, 0, UD * sizeof(float), stream);
    spmm_atomic_kernel<<<R_blk, TPB, 0, stream>>>(R_rows, R_cols, R_vals, itA, text_ui, NNZ_R);
    copy_kernel<<<blocks_for(ID, TPB), TPB, 0, stream>>>(itA, text_ui + UD, ID);

    // ---- attention scores: h = leaky(e @ W_common1 + b), s = h . w_common2 ----
    {
        int waves = ((NTOT + 15) / 16) * (D / 16);
        int blk = blocks_for((size_t)waves * 32, TPB);
        int rd_blk = blocks_for((size_t)NTOT * 32, TPB);

        gemm64_wmma_kernel<<<blk, TPB, 0, stream>>>(
            image_ui, W_common1, b_common1, nullptr, nullptr, nullptr, hbuf, NTOT, D, 1);
        rowdot_kernel<<<rd_blk, TPB, 0, stream>>>(hbuf, w_common2, s_img, NTOT);

        gemm64_wmma_kernel<<<blk, TPB, 0, stream>>>(
            text_ui, W_common1, b_common1, nullptr, nullptr, nullptr, hbuf, NTOT, D, 1);
        rowdot_kernel<<<rd_blk, TPB, 0, stream>>>(hbuf, w_common2, s_txt, NTOT);
    }

    // ---- softmax fuse -> out[1]=special_img, out[2]=special_txt, out[3]=common ----
    fuse_kernel<<<blocks_for(ND, TPB), TPB, 0, stream>>>(
        image_ui, text_ui, s_img, s_txt,
        out + ND * 1, out + ND * 2, out + ND * 3, ND);
}